// VNETDetector_89069031784866
// MI455X (gfx1250) — compile-verified
//
#include <hip/hip_runtime.h>

typedef float v2f __attribute__((ext_vector_type(2)));
typedef float v8f __attribute__((ext_vector_type(8)));

#define HUGEF 1e30f

__device__ __forceinline__ float relu_(float x) { return fmaxf(x, 0.0f); }

// ---------------------------------------------------------------------------
// Kernel 1: priors[t][n] = log_softmax_n( relu(rx[t]*W1 + b1) @ W2 + b2 )
// One wave32 computes a 16(timestep) x 16(state) tile with
// V_WMMA_F32_16X16X4_F32.  K is padded with zeros to 80 in LDS, so the K-loop
// is a compile-time fully-unrolled 20 WMMA chain (valid for any H <= 80;
// reference has H = 75).  Epilogue: LDS transpose (padded rows), per-row
// register log_softmax, coalesced b128 stores.
// ---------------------------------------------------------------------------
__global__ void __launch_bounds__(256) priors_wmma(
    const float* __restrict__ rx, const float* __restrict__ W1,
    const float* __restrict__ b1, const float* __restrict__ W2,
    const float* __restrict__ b2, float* __restrict__ priors,
    int T, int H)
{
  __shared__ float sW1[80], sb1[80], sW2[80 * 16], sb2[16];
  __shared__ __align__(16) float stile[8][16 * 20];   // per-wave tile, row stride 20
  const int tid = threadIdx.x;
  for (int i = tid; i < 80; i += blockDim.x) {
    sW1[i] = (i < H) ? W1[i] : 0.0f;
    sb1[i] = (i < H) ? b1[i] : 0.0f;
  }
  for (int i = tid; i < 80 * 16; i += blockDim.x) {
    sW2[i] = ((i >> 4) < H) ? W2[i] : 0.0f;
  }
  if (tid < 16) sb2[tid] = b2[tid];
  __syncthreads();

  const int lane = tid & 31;
  const int wv = blockIdx.x * (blockDim.x >> 5) + (tid >> 5);
  const int base = wv * 16;
  if (base >= T) return;

  const int n  = lane & 15;
  const int hi = lane >> 4;
  const int tr = base + n;
  const float rxv = (tr < T) ? rx[tr] : 0.0f;

  v8f acc = {};
#pragma unroll
  for (int k0 = 0; k0 < 80; k0 += 4) {    // compile-time: 20 WMMAs, pipelined
    const int ka = k0 + (hi << 1);
    v2f A, B;
    A.x = relu_(fmaf(rxv, sW1[ka],     sb1[ka]));
    A.y = relu_(fmaf(rxv, sW1[ka + 1], sb1[ka + 1]));
    B.x = sW2[(ka << 4) + n];
    B.y = sW2[((ka + 1) << 4) + n];
    acc = __builtin_amdgcn_wmma_f32_16x16x4_f32(
        false, A, false, B, (short)0, acc, false, false);
  }

  // scatter tile (plus b2) into padded LDS: row i+(hi*8), col n
  const float lb2 = sb2[n];
  float* tw = stile[tid >> 5];
#pragma unroll
  for (int i = 0; i < 8; ++i)
    tw[(i + (hi << 3)) * 20 + n] = acc[i] + lb2;
  __builtin_amdgcn_wave_barrier();   // LDS is in-order within a wave

  if (lane < 16) {
    const int t = base + lane;
    if (t < T) {
      const float* rp = &tw[lane * 20];
      float x[16];
#pragma unroll
      for (int i = 0; i < 4; ++i) {
        const float4 q = ((const float4*)rp)[i];
        x[4*i + 0] = q.x; x[4*i + 1] = q.y; x[4*i + 2] = q.z; x[4*i + 3] = q.w;
      }
      float mx = x[0];
#pragma unroll
      for (int k = 1; k < 16; ++k) mx = fmaxf(mx, x[k]);
      float se = 0.0f;
#pragma unroll
      for (int k = 0; k < 16; ++k) se += __expf(x[k] - mx);
      const float lse = mx + __logf(se);
      float4* op = (float4*)(priors + (size_t)t * 16);
#pragma unroll
      for (int i = 0; i < 4; ++i)
        op[i] = make_float4(x[4*i] - lse, x[4*i + 1] - lse,
                            x[4*i + 2] - lse, x[4*i + 3] - lse);
    }
  }
}

// ---------------------------------------------------------------------------
// Kernel 2 (pass 1): chunk min-plus transition matrices, register-only.
// Lane = matrix row j (lane&15); each 16-lane half of a wave owns one chunk.
// ---------------------------------------------------------------------------
__global__ void __launch_bounds__(256) viterbi_pass1(
    const float* __restrict__ priors, float* __restrict__ mats,
    int T, int clen, int nchunk)
{
  const int lane = threadIdx.x & 31;
  const int wv = blockIdx.x * (blockDim.x >> 5) + (threadIdx.x >> 5);
  const int j = lane & 15;                    // matrix row owned by lane
  const int c = 2 * wv + (lane >> 4);         // chunk per 16-lane half
  if (c >= nchunk) return;                    // wave-uniform (nchunk even)

  float a[16];
#pragma unroll
  for (int s = 0; s < 16; ++s) a[s] = (s == j) ? 0.0f : HUGEF;

  const int start = c * clen;
  const int end   = min(start + clen, T);
  const float4* pp = (const float4*)priors;   // row t = pp[4t .. 4t+3]

  for (int t = start; t < end; ++t) {
    __builtin_prefetch(&pp[(size_t)(t + 16) * 4], 0, 1);
    const float4 q0 = pp[(size_t)t * 4 + 0];
    const float4 q1 = pp[(size_t)t * 4 + 1];
    const float4 q2 = pp[(size_t)t * 4 + 2];
    const float4 q3 = pp[(size_t)t * 4 + 3];
    const float p[16] = { q0.x,q0.y,q0.z,q0.w, q1.x,q1.y,q1.z,q1.w,
                          q2.x,q2.y,q2.z,q2.w, q3.x,q3.y,q3.z,q3.w };
    float u[16];
#pragma unroll
    for (int s = 0; s < 16; ++s) u[s] = a[s] - p[s];
#pragma unroll
    for (int s = 0; s < 16; ++s)
      a[s] = fminf(u[(2 * s) & 15], u[(2 * s + 1) & 15]);
  }

  float4* mrow = (float4*)(mats + (size_t)c * 256 + j * 16);
  mrow[0] = make_float4(a[0],  a[1],  a[2],  a[3]);
  mrow[1] = make_float4(a[4],  a[5],  a[6],  a[7]);
  mrow[2] = make_float4(a[8],  a[9],  a[10], a[11]);
  mrow[3] = make_float4(a[12], a[13], a[14], a[15]);
}

// ---------------------------------------------------------------------------
// Kernel 3 (combine): one wave propagates the entry metric vector through all
// chunk matrices:  v'[s] = min_j ( v[j] + M[j][s] ).
// ---------------------------------------------------------------------------
__global__ void viterbi_combine(const float* __restrict__ mats,
                                float* __restrict__ vecs, int nchunk)
{
  const int lane = threadIdx.x & 31;
  const int s  = lane & 15;
  const int hf = lane >> 4;
  float v = 0.0f;
  for (int c = 0; c < nchunk; ++c) {
    __builtin_prefetch(mats + (size_t)(c + 2) * 256 + (hf << 7) + s, 0, 1);
    if (lane < 16) vecs[(size_t)c * 16 + s] = v;
    const float* M = mats + (size_t)c * 256;
    float acc = HUGEF;
#pragma unroll
    for (int j = 0; j < 8; ++j) {
      const int row = (hf << 3) + j;
      acc = fminf(acc, __shfl(v, row, 16) + M[row * 16 + s]);
    }
    v = fminf(acc, __shfl_xor(acc, 16, 32));
  }
}

// ---------------------------------------------------------------------------
// Kernel 4 (pass 2): one chunk per LANE, whole 16-state metric vector in
// registers.  Zero cross-lane ops per step.
// ---------------------------------------------------------------------------
__global__ void __launch_bounds__(256) viterbi_pass2(
    const float* __restrict__ priors, const float* __restrict__ vecs,
    float* __restrict__ out, int T, int clen, int nchunk)
{
  const int lane = threadIdx.x & 31;
  const int wv = blockIdx.x * (blockDim.x >> 5) + (threadIdx.x >> 5);
  const int c = wv * 32 + lane;
  if (c >= nchunk) return;                 // wave-uniform (nchunk % 32 == 0)

  float v[16];
  {
    const float4* vv = (const float4*)(vecs + (size_t)c * 16);
    const float4 r0 = vv[0], r1 = vv[1], r2 = vv[2], r3 = vv[3];
    v[0]=r0.x; v[1]=r0.y; v[2]=r0.z; v[3]=r0.w;
    v[4]=r1.x; v[5]=r1.y; v[6]=r1.z; v[7]=r1.w;
    v[8]=r2.x; v[9]=r2.y; v[10]=r2.z; v[11]=r2.w;
    v[12]=r3.x; v[13]=r3.y; v[14]=r3.z; v[15]=r3.w;
  }

  const float4* pp = (const float4*)priors;
  const int start = c * clen;
  const int end   = min(start + clen, T);

  for (int t = start; t < end; ++t) {
    __builtin_prefetch(&pp[(size_t)(t + 16) * 4], 0, 1);
    const float4 q0 = pp[(size_t)t * 4 + 0];
    const float4 q1 = pp[(size_t)t * 4 + 1];
    const float4 q2 = pp[(size_t)t * 4 + 2];
    const float4 q3 = pp[(size_t)t * 4 + 3];
    const float p[16] = { q0.x,q0.y,q0.z,q0.w, q1.x,q1.y,q1.z,q1.w,
                          q2.x,q2.y,q2.z,q2.w, q3.x,q3.y,q3.z,q3.w };

    // argmin with first-index tie-break: left-wins-ties register tree
    float bv[8]; int bi[8];
#pragma unroll
    for (int i = 0; i < 8; ++i) {
      const bool tk = v[2*i + 1] < v[2*i];
      bv[i] = tk ? v[2*i + 1] : v[2*i];
      bi[i] = 2*i + (tk ? 1 : 0);
    }
#pragma unroll
    for (int i = 0; i < 4; ++i) {
      const bool tk = bv[2*i + 1] < bv[2*i];
      bv[i] = tk ? bv[2*i + 1] : bv[2*i];
      bi[i] = tk ? bi[2*i + 1] : bi[2*i];
    }
    {
      bool tk = bv[1] < bv[0];
      bv[0] = tk ? bv[1] : bv[0];  bi[0] = tk ? bi[1] : bi[0];
      tk = bv[3] < bv[2];
      bv[2] = tk ? bv[3] : bv[2];  bi[2] = tk ? bi[3] : bi[2];
      tk = bv[2] < bv[0];
      bv[0] = tk ? bv[2] : bv[0];  bi[0] = tk ? bi[2] : bi[0];
    }
    const float m = bv[0];

    float e = 0.0f;
#pragma unroll
    for (int s = 0; s < 16; ++s) e += __expf(m - v[s]);

    out[t]             = (float)(bi[0] & 1);
    out[(size_t)T + t] = 2.0f / e;

    float u[16];
#pragma unroll
    for (int s = 0; s < 16; ++s) u[s] = v[s] - p[s];
#pragma unroll
    for (int s = 0; s < 16; ++s)
      v[s] = fminf(u[(2 * s) & 15], u[(2 * s + 1) & 15]);
  }
}

// ---------------------------------------------------------------------------
extern "C" void kernel_launch(void* const* d_in, const int* in_sizes, int n_in,
                              void* d_out, int out_size, void* d_ws, size_t ws_size,
                              hipStream_t stream)
{
  (void)n_in; (void)out_size; (void)ws_size;
  const float* rx = (const float*)d_in[0];
  const float* W1 = (const float*)d_in[1];
  const float* b1 = (const float*)d_in[2];
  const float* W2 = (const float*)d_in[3];
  const float* b2 = (const float*)d_in[4];
  float* out = (float*)d_out;
  const int T = in_sizes[0];   // rx is [T,1]
  const int H = in_sizes[1];   // W1 is [1,H]

  int nchunk = 1536;                       // multiple of 32, even
  while (nchunk > 32 && (size_t)nchunk * 16 > (size_t)T) nchunk >>= 1;
  const int clen = (T + nchunk - 1) / nchunk;

  // workspace: priors [T*16] | chunk matrices [nchunk*256] | vecs [nchunk*16]
  float* priors = (float*)d_ws;
  float* mats   = priors + (size_t)T * 16;
  float* vecs   = mats + (size_t)nchunk * 256;

  const int tiles = (T + 15) / 16;
  dim3 blk(256);
  priors_wmma   <<<(tiles + 7) / 8, blk, 0, stream>>>(rx, W1, b1, W2, b2, priors, T, H);
  viterbi_pass1 <<<((nchunk / 2) + 7) / 8, blk, 0, stream>>>(priors, mats, T, clen, nchunk);
  viterbi_combine<<<1, 32, 0, stream>>>(mats, vecs, nchunk);
  viterbi_pass2 <<<(nchunk + 255) / 256, blk, 0, stream>>>(priors, vecs, out, T, clen, nchunk);
}